// Model02_54331336294858
// MI455X (gfx1250) — compile-verified
//
#include <hip/hip_runtime.h>
#include <hip/hip_bf16.h>

#define GRU_B 1024
#define GRU_T 4096
#define GRU_D 8
#define TCHUNK 64

typedef float v2f __attribute__((ext_vector_type(2)));
typedef float v8f __attribute__((ext_vector_type(8)));

// ---------------------------------------------------------------------------
// Fast tanh: CDNA5 has a TRANS v_tanh_f32. Fall back to exp2+rcp otherwise.
// ---------------------------------------------------------------------------
__device__ __forceinline__ float tanh_fast(float x) {
#if __has_builtin(__builtin_amdgcn_tanhf)
  return __builtin_amdgcn_tanhf(x);
#else
  // tanh(x) = 2/(1+exp(-2x)) - 1 ; exp(-2x) = exp2(-2*log2(e)*x)
  float e = __builtin_amdgcn_exp2f(x * -2.885390081777927f);
  return __builtin_fmaf(2.0f, __builtin_amdgcn_rcpf(1.0f + e), -1.0f);
#endif
}

// ---------------------------------------------------------------------------
// Kernel 1: layer-0 input projection via V_WMMA_F32_16X16X4_F32.
//   pre[0][t][b] = 0.5*(x·w_r + b_ih_r + b_hh_r)   (half-scaled for tanh-sigmoid)
//   pre[1][t][b] = 0.5*(x·w_z + b_ih_z + b_hh_z)
//   pre[2][t][b] =      x·w_n + b_ih_n
// Tile: M = 3 gates (A = scaled weights, 16x4), N = 16 consecutive b
// (B operand = x rows, 4x16), K = 8 split into two K=4 WMMAs chained via C.
// Scale+bias folded into A and the C accumulator.
// ---------------------------------------------------------------------------
__global__ void gru0_pre_wmma(const float* __restrict__ x,
                              const float* __restrict__ w_ih0,
                              const float* __restrict__ b_ih0,
                              const float* __restrict__ b_hh0,
                              float* __restrict__ pre) {
  const int lane = threadIdx.x & 31;
  const int m    = lane & 15;   // A: gate row / B: batch column within tile
  const int kh   = lane >> 4;   // 0 -> K=0,1 ; 1 -> K=2,3 (per 32-bit A/B layout)
  const int b0   = blockIdx.x * 16;
  const int t0   = blockIdx.y * TCHUNK;
  const size_t TB = (size_t)GRU_T * GRU_B;

#if __has_builtin(__builtin_amdgcn_wmma_f32_16x16x4_f32)
  v2f a_lo = {0.0f, 0.0f}, a_hi = {0.0f, 0.0f};
  if (m < 3) {
    const float sc = (m == 2) ? 1.0f : 0.5f;   // r,z halved; n full scale
    const int k0 = kh * 2;
    a_lo.x = sc * w_ih0[m * GRU_D + k0];
    a_lo.y = sc * w_ih0[m * GRU_D + k0 + 1];
    a_hi.x = sc * w_ih0[m * GRU_D + k0 + 4];
    a_hi.y = sc * w_ih0[m * GRU_D + k0 + 5];
  }
  // C bias: D rows 0..7 live in lanes 0-15 (vgpr v = row v); rows 8-15 in 16-31.
  v8f cb = {};
  if (lane < 16) {
    cb[0] = 0.5f * (b_ih0[0] + b_hh0[0]);
    cb[1] = 0.5f * (b_ih0[1] + b_hh0[1]);
    cb[2] = b_ih0[2];
  }
  const float* xr = x + (size_t)(b0 + m) * GRU_T * GRU_D + kh * 2;
  for (int tt = 0; tt < TCHUNK; ++tt) {
    const int t = t0 + tt;
    const float* px = xr + (size_t)t * GRU_D;
    v2f bl, bh;
    bl.x = px[0]; bl.y = px[1];   // K = kh*2, kh*2+1
    bh.x = px[4]; bh.y = px[5];   // K = kh*2+4, kh*2+5
    v8f acc = __builtin_amdgcn_wmma_f32_16x16x4_f32(
        false, a_lo, false, bl, (short)0, cb, false, false);
    acc = __builtin_amdgcn_wmma_f32_16x16x4_f32(
        false, a_hi, false, bh, (short)0, acc, false, false);
    if (lane < 16) {  // lanes 0-15 hold N = b0+lane, rows (gates) 0..2 in vgprs 0..2
      const size_t o = (size_t)t * GRU_B + (size_t)(b0 + lane);
      pre[o]          = acc[0];
      pre[TB + o]     = acc[1];
      pre[2 * TB + o] = acc[2];
    }
  }
#else
  // Scalar fallback (should not trigger; WMMA builtin is probe-confirmed).
  if (lane < 16) {
    const int b = b0 + lane;
    float wr[GRU_D], wz[GRU_D], wn[GRU_D];
    #pragma unroll
    for (int k = 0; k < GRU_D; ++k) {
      wr[k] = 0.5f * w_ih0[k];
      wz[k] = 0.5f * w_ih0[GRU_D + k];
      wn[k] = w_ih0[2 * GRU_D + k];
    }
    const float cr = 0.5f * (b_ih0[0] + b_hh0[0]);
    const float cz = 0.5f * (b_ih0[1] + b_hh0[1]);
    const float cn = b_ih0[2];
    for (int tt = 0; tt < TCHUNK; ++tt) {
      const int t = t0 + tt;
      const float* px = x + ((size_t)b * GRU_T + t) * GRU_D;
      float ar = cr, az = cz, an = cn;
      #pragma unroll
      for (int k = 0; k < GRU_D; ++k) {
        ar = __builtin_fmaf(px[k], wr[k], ar);
        az = __builtin_fmaf(px[k], wz[k], az);
        an = __builtin_fmaf(px[k], wn[k], an);
      }
      const size_t o = (size_t)t * GRU_B + b;
      pre[o] = ar; pre[TB + o] = az; pre[2 * TB + o] = an;
    }
  }
#endif
}

// ---------------------------------------------------------------------------
// Recurrence constants (everything precomputable once per thread).
// Suffix h = halved (for sigmoid-as-tanh argument).
// ---------------------------------------------------------------------------
struct RecConsts {
  float whr0h, whz0h, whn0, bhn0;                                   // layer 0
  float wir1h, wiz1h, win1, bin1, whr1h, whz1h, whn1, bhn1, crb1, czb1; // layer 1
  float wir2h, wiz2h, win2, bin2, whr2h, whz2h, whn2, bhn2, crb2, czb2; // layer 2
};

__device__ __forceinline__ RecConsts load_consts(
    const float* w_hh0, const float* b_hh0,
    const float* w_ih1, const float* w_hh1, const float* b_ih1, const float* b_hh1,
    const float* w_ih2, const float* w_hh2, const float* b_ih2, const float* b_hh2) {
  RecConsts c;
  c.whr0h = 0.5f * w_hh0[0]; c.whz0h = 0.5f * w_hh0[1];
  c.whn0  = w_hh0[2];        c.bhn0  = b_hh0[2];

  c.wir1h = 0.5f * w_ih1[0]; c.wiz1h = 0.5f * w_ih1[1];
  c.win1  = w_ih1[2];        c.bin1  = b_ih1[2];
  c.whr1h = 0.5f * w_hh1[0]; c.whz1h = 0.5f * w_hh1[1];
  c.whn1  = w_hh1[2];        c.bhn1  = b_hh1[2];
  c.crb1  = 0.5f * (b_ih1[0] + b_hh1[0]);
  c.czb1  = 0.5f * (b_ih1[1] + b_hh1[1]);

  c.wir2h = 0.5f * w_ih2[0]; c.wiz2h = 0.5f * w_ih2[1];
  c.win2  = w_ih2[2];        c.bin2  = b_ih2[2];
  c.whr2h = 0.5f * w_hh2[0]; c.whz2h = 0.5f * w_hh2[1];
  c.whn2  = w_hh2[2];        c.bhn2  = b_hh2[2];
  c.crb2  = 0.5f * (b_ih2[0] + b_hh2[0]);
  c.czb2  = 0.5f * (b_ih2[1] + b_hh2[1]);
  return c;
}

// One fused time-step across all three layers (latency-optimized).
// ar_p/az_p are half-scaled pre-activations incl. both biases; gn_p incl. b_ih_n.
__device__ __forceinline__ void gru_step(const RecConsts& c,
                                         float ar_p, float az_p, float gn_p,
                                         float& h0, float& h1, float& h2) {
  // ---- layer 0 ----
  float ar = __builtin_fmaf(h0, c.whr0h, ar_p);
  float az = __builtin_fmaf(h0, c.whz0h, az_p);
  float gh = __builtin_fmaf(h0, c.whn0, c.bhn0);
  float rt = tanh_fast(ar);            // tanh(0.5*a_r)
  float zt = tanh_fast(az);
  float p  = 0.5f * gh;
  float an = __builtin_fmaf(rt, p, gn_p + p);   // gxn + sigmoid(a_r)*ghn
  float nn = tanh_fast(an);
  float z  = __builtin_fmaf(0.5f, zt, 0.5f);
  h0 = __builtin_fmaf(z, h0 - nn, nn);          // (1-z)*n + z*h

  // ---- layer 1 (h1-dependent terms are off the h0-chain) ----
  float c1r = __builtin_fmaf(h1, c.whr1h, c.crb1);
  float c1z = __builtin_fmaf(h1, c.whz1h, c.czb1);
  float gh1 = __builtin_fmaf(h1, c.whn1, c.bhn1);
  float ar1 = __builtin_fmaf(h0, c.wir1h, c1r);
  float az1 = __builtin_fmaf(h0, c.wiz1h, c1z);
  float gx1 = __builtin_fmaf(h0, c.win1, c.bin1);
  float rt1 = tanh_fast(ar1);
  float zt1 = tanh_fast(az1);
  float p1  = 0.5f * gh1;
  float an1 = __builtin_fmaf(rt1, p1, gx1 + p1);
  float n1  = tanh_fast(an1);
  float z1  = __builtin_fmaf(0.5f, zt1, 0.5f);
  h1 = __builtin_fmaf(z1, h1 - n1, n1);

  // ---- layer 2 ----
  float c2r = __builtin_fmaf(h2, c.whr2h, c.crb2);
  float c2z = __builtin_fmaf(h2, c.whz2h, c.czb2);
  float gh2 = __builtin_fmaf(h2, c.whn2, c.bhn2);
  float ar2 = __builtin_fmaf(h1, c.wir2h, c2r);
  float az2 = __builtin_fmaf(h1, c.wiz2h, c2z);
  float gx2 = __builtin_fmaf(h1, c.win2, c.bin2);
  float rt2 = tanh_fast(ar2);
  float zt2 = tanh_fast(az2);
  float p2  = 0.5f * gh2;
  float an2 = __builtin_fmaf(rt2, p2, gx2 + p2);
  float n2  = tanh_fast(an2);
  float z2  = __builtin_fmaf(0.5f, zt2, 0.5f);
  h2 = __builtin_fmaf(z2, h2 - n2, n2);
}

// ---------------------------------------------------------------------------
// Kernel 2: the sequential scan. One thread per batch element.
// 4-deep register pipeline + global_prefetch keeps memory off the chain
// (the 48MB pre array lives in the 192MB L2 anyway).
// ---------------------------------------------------------------------------
__global__ void gru_rec(const float* __restrict__ pre,
                        const float* __restrict__ w_hh0, const float* __restrict__ b_hh0,
                        const float* __restrict__ w_ih1, const float* __restrict__ w_hh1,
                        const float* __restrict__ b_ih1, const float* __restrict__ b_hh1,
                        const float* __restrict__ w_ih2, const float* __restrict__ w_hh2,
                        const float* __restrict__ b_ih2, const float* __restrict__ b_hh2,
                        float* __restrict__ out) {
  const int b = blockIdx.x * blockDim.x + threadIdx.x;
  if (b >= GRU_B) return;
  const size_t TB = (size_t)GRU_T * GRU_B;
  const float* pr = pre;
  const float* pz = pre + TB;
  const float* pn = pre + 2 * TB;

  const RecConsts c = load_consts(w_hh0, b_hh0, w_ih1, w_hh1, b_ih1, b_hh1,
                                  w_ih2, w_hh2, b_ih2, b_hh2);

  float h0 = 0.0f, h1 = 0.0f, h2 = 0.0f;
  float arb[4], azb[4], gnb[4];
  #pragma unroll
  for (int i = 0; i < 4; ++i) {
    const size_t o = (size_t)i * GRU_B + b;
    arb[i] = pr[o]; azb[i] = pz[o]; gnb[i] = pn[o];
  }

  #pragma unroll 4
  for (int t = 0; t < GRU_T; ++t) {
    const int s = t & 3;
    const float ar_p = arb[s], az_p = azb[s], gn_p = gnb[s];
    int tp = t + 4; if (tp >= GRU_T) tp = GRU_T - 1;
    const size_t o = (size_t)tp * GRU_B + b;
    arb[s] = pr[o]; azb[s] = pz[o]; gnb[s] = pn[o];
    if ((t & 15) == 0) {
      const int tf = (t + 32 < GRU_T) ? (t + 32) : t;
      const size_t po = (size_t)tf * GRU_B + b;
      __builtin_prefetch(&pr[po], 0, 1);
      __builtin_prefetch(&pz[po], 0, 1);
      __builtin_prefetch(&pn[po], 0, 1);
    }
    gru_step(c, ar_p, az_p, gn_p, h0, h1, h2);
  }
  out[b] = h2;
}

// ---------------------------------------------------------------------------
// Fallback: fused single kernel (no workspace). Layer-0 dot computed inline
// (off the dependency chain).
// ---------------------------------------------------------------------------
__global__ void gru_rec_fused(const float* __restrict__ x,
                              const float* __restrict__ w_ih0, const float* __restrict__ w_hh0,
                              const float* __restrict__ b_ih0, const float* __restrict__ b_hh0,
                              const float* __restrict__ w_ih1, const float* __restrict__ w_hh1,
                              const float* __restrict__ b_ih1, const float* __restrict__ b_hh1,
                              const float* __restrict__ w_ih2, const float* __restrict__ w_hh2,
                              const float* __restrict__ b_ih2, const float* __restrict__ b_hh2,
                              float* __restrict__ out) {
  const int b = blockIdx.x * blockDim.x + threadIdx.x;
  if (b >= GRU_B) return;
  float wr[GRU_D], wz[GRU_D], wn[GRU_D];
  #pragma unroll
  for (int k = 0; k < GRU_D; ++k) {
    wr[k] = 0.5f * w_ih0[k];
    wz[k] = 0.5f * w_ih0[GRU_D + k];
    wn[k] = w_ih0[2 * GRU_D + k];
  }
  const float cr0 = 0.5f * (b_ih0[0] + b_hh0[0]);
  const float cz0 = 0.5f * (b_ih0[1] + b_hh0[1]);
  const float cn0 = b_ih0[2];
  const RecConsts c = load_consts(w_hh0, b_hh0, w_ih1, w_hh1, b_ih1, b_hh1,
                                  w_ih2, w_hh2, b_ih2, b_hh2);
  const float* px = x + (size_t)b * GRU_T * GRU_D;
  float h0 = 0.0f, h1 = 0.0f, h2 = 0.0f;
  for (int t = 0; t < GRU_T; ++t) {
    float xv[GRU_D];
    #pragma unroll
    for (int k = 0; k < GRU_D; ++k) xv[k] = px[(size_t)t * GRU_D + k];
    float ar_p = cr0, az_p = cz0, gn_p = cn0;
    #pragma unroll
    for (int k = 0; k < GRU_D; ++k) {
      ar_p = __builtin_fmaf(xv[k], wr[k], ar_p);
      az_p = __builtin_fmaf(xv[k], wz[k], az_p);
      gn_p = __builtin_fmaf(xv[k], wn[k], gn_p);
    }
    gru_step(c, ar_p, az_p, gn_p, h0, h1, h2);
  }
  out[b] = h2;
}

extern "C" void kernel_launch(void* const* d_in, const int* in_sizes, int n_in,
                              void* d_out, int out_size, void* d_ws, size_t ws_size,
                              hipStream_t stream) {
  const float* x     = (const float*)d_in[0];
  const float* w_ih0 = (const float*)d_in[1];
  const float* w_hh0 = (const float*)d_in[2];
  const float* b_ih0 = (const float*)d_in[3];
  const float* b_hh0 = (const float*)d_in[4];
  const float* w_ih1 = (const float*)d_in[5];
  const float* w_hh1 = (const float*)d_in[6];
  const float* b_ih1 = (const float*)d_in[7];
  const float* b_hh1 = (const float*)d_in[8];
  const float* w_ih2 = (const float*)d_in[9];
  const float* w_hh2 = (const float*)d_in[10];
  const float* b_ih2 = (const float*)d_in[11];
  const float* b_hh2 = (const float*)d_in[12];
  float* out = (float*)d_out;

  const size_t need = 3ull * GRU_T * GRU_B * sizeof(float);  // 48 MB
  if (ws_size >= need) {
    float* pre = (float*)d_ws;
    dim3 grid(GRU_B / 16, GRU_T / TCHUNK);
    gru0_pre_wmma<<<grid, 32, 0, stream>>>(x, w_ih0, b_ih0, b_hh0, pre);
    gru_rec<<<GRU_B / 64, 64, 0, stream>>>(pre, w_hh0, b_hh0,
                                           w_ih1, w_hh1, b_ih1, b_hh1,
                                           w_ih2, w_hh2, b_ih2, b_hh2, out);
  } else {
    gru_rec_fused<<<GRU_B / 64, 64, 0, stream>>>(x, w_ih0, w_hh0, b_ih0, b_hh0,
                                                 w_ih1, w_hh1, b_ih1, b_hh1,
                                                 w_ih2, w_hh2, b_ih2, b_hh2, out);
  }
}